// TorchEAM_42485816492264
// MI455X (gfx1250) — compile-verified
//
#include <hip/hip_runtime.h>

// ---------------------------------------------------------------------------
// EAM energy + forces for gfx1250 (MI455X). 3-pass scatter/gather.
// No matrix ops in this workload -> WMMA inapplicable; the kernel is bound by
// the L2/atomic path. Spline tables are repacked into 32B AoS records so each
// edge does two global_load_b128 from one cacheline instead of 8 scattered
// b32 loads. All scatter targets (node_rho 400KB, forces 1.2MB) and tables
// (480KB packed) are L2-resident (192MB L2).
// ---------------------------------------------------------------------------

#define NR_K    10000
#define NRHO_K  10000
#define CUTOFF_F  6.0f
#define RHOMAX_F 60.0f

__device__ __forceinline__ float atomic_add_f32(float* p, float v) {
    // Relaxed agent-scope fp32 atomic -> hardware global_atomic_add_f32.
    return __hip_atomic_fetch_add(p, v, __ATOMIC_RELAXED, __HIP_MEMORY_SCOPE_AGENT);
}

__device__ __forceinline__ float wave_sum(float v) {
    // wave32 butterfly reduction (gfx1250 is wave32-only)
    v += __shfl_xor(v, 16, 32);
    v += __shfl_xor(v,  8, 32);
    v += __shfl_xor(v,  4, 32);
    v += __shfl_xor(v,  2, 32);
    v += __shfl_xor(v,  1, 32);
    return v;
}

// ======================= packed-table kernels (fast path) ===================
// rad_pack[2*i+0] = (a0,a1,b0,b1), rad_pack[2*i+1] = (c0,c1,d0,d1)
// emb_pack[i]     = (a,b,c,d)

__global__ __launch_bounds__(256) void eam_pass1_packed(
    const float* __restrict__ r, const int* __restrict__ dst,
    const float* __restrict__ rs, const float4* __restrict__ rad_pack,
    float* __restrict__ node_rho, float* __restrict__ E, int n_edges)
{
    const float inv_h = (float)(NR_K - 1) / CUTOFF_F;
    const int stride = gridDim.x * blockDim.x;
    const float3* __restrict__ r3 = reinterpret_cast<const float3*>(r);

    float phi_acc = 0.0f;
    for (int e = blockIdx.x * blockDim.x + threadIdx.x; e < n_edges; e += stride) {
        if (e + stride < n_edges)
            __builtin_prefetch(&r3[e + stride], 0, 1);   // global_prefetch_b8

        float3 v = r3[e];
        float bl = sqrtf(fmaf(v.x, v.x, fmaf(v.y, v.y, v.z * v.z)));

        int idx = (int)(bl * inv_h);
        idx = min(idx, NR_K - 2);               // bl >= 0 always
        float s = bl - rs[idx];

        float4 p0 = rad_pack[2 * idx];          // global_load_b128
        float4 p1 = rad_pack[2 * idx + 1];      // global_load_b128 (same line)

        float rho  = fmaf(s, fmaf(s, fmaf(s, p1.z, p1.x), p0.z), p0.x);
        float rphi = fmaf(s, fmaf(s, fmaf(s, p1.w, p1.y), p0.w), p0.y);

        atomic_add_f32(&node_rho[dst[e]], rho);     // L2-resident scatter
        phi_acc += 0.5f * rphi / bl;
    }
    float w = wave_sum(phi_acc);
    if ((threadIdx.x & 31) == 0) atomic_add_f32(E, w);
}

__global__ __launch_bounds__(256) void eam_pass2_packed(
    const float* __restrict__ node_rho, const float* __restrict__ rhos,
    const float4* __restrict__ emb_pack,
    float* __restrict__ Fprime, float* __restrict__ E, int n_nodes)
{
    const float inv_h = (float)(NRHO_K - 1) / RHOMAX_F;
    int i = blockIdx.x * blockDim.x + threadIdx.x;

    float Facc = 0.0f;
    if (i < n_nodes) {
        float rho = node_rho[i];
        int idx = (int)(rho * inv_h);
        idx = max(0, min(idx, NRHO_K - 2));
        float s = rho - rhos[idx];

        float4 p = emb_pack[idx];               // (a,b,c,d) one b128
        Facc      = fmaf(s, fmaf(s, fmaf(s, p.w, p.z), p.y), p.x);
        Fprime[i] = fmaf(s, fmaf(s, 3.0f * p.w, 2.0f * p.z), p.y);
    }
    float w = wave_sum(Facc);
    if ((threadIdx.x & 31) == 0) atomic_add_f32(E, w);
}

__global__ __launch_bounds__(256) void eam_pass3_packed(
    const float* __restrict__ r, const int* __restrict__ src, const int* __restrict__ dst,
    const float* __restrict__ rs, const float4* __restrict__ rad_pack,
    const float* __restrict__ Fprime, float* __restrict__ forces, int n_edges)
{
    const float inv_h = (float)(NR_K - 1) / CUTOFF_F;
    const int stride = gridDim.x * blockDim.x;
    const float3* __restrict__ r3 = reinterpret_cast<const float3*>(r);

    for (int e = blockIdx.x * blockDim.x + threadIdx.x; e < n_edges; e += stride) {
        if (e + stride < n_edges)
            __builtin_prefetch(&r3[e + stride], 0, 1);

        float3 v = r3[e];
        float bl = sqrtf(fmaf(v.x, v.x, fmaf(v.y, v.y, v.z * v.z)));

        int idx = (int)(bl * inv_h);
        idx = min(idx, NR_K - 2);
        float s = bl - rs[idx];

        float4 p0 = rad_pack[2 * idx];          // (a0,a1,b0,b1)
        float4 p1 = rad_pack[2 * idx + 1];      // (c0,c1,d0,d1)

        float drho  = fmaf(s, fmaf(s, 3.0f * p1.z, 2.0f * p1.x), p0.z);
        float rphi  = fmaf(s, fmaf(s, fmaf(s, p1.w, p1.y), p0.w), p0.y);
        float rphip = fmaf(s, fmaf(s, 3.0f * p1.w, 2.0f * p1.y), p0.w);

        float inv_bl = 1.0f / bl;
        float dphi = (rphip - rphi * inv_bl) * inv_bl;   // d/dbl (rphi/bl)

        int de = dst[e], se = src[e];
        float g = fmaf(Fprime[de], drho, 0.5f * dphi);
        float scale = g * inv_bl;

        float fx = scale * v.x, fy = scale * v.y, fz = scale * v.z;
        atomic_add_f32(&forces[3 * se + 0],  fx);
        atomic_add_f32(&forces[3 * se + 1],  fy);
        atomic_add_f32(&forces[3 * se + 2],  fz);
        atomic_add_f32(&forces[3 * de + 0], -fx);
        atomic_add_f32(&forces[3 * de + 1], -fy);
        atomic_add_f32(&forces[3 * de + 2], -fz);
    }
}

// Repack SoA spline coeffs into cacheline-friendly AoS records (per call;
// deterministic, no cached state).
__global__ __launch_bounds__(256) void eam_repack(
    const float* __restrict__ rad_a, const float* __restrict__ rad_b,
    const float* __restrict__ rad_c, const float* __restrict__ rad_d,
    const float* __restrict__ emb_a, const float* __restrict__ emb_b,
    const float* __restrict__ emb_c, const float* __restrict__ emb_d,
    float4* __restrict__ rad_pack, float4* __restrict__ emb_pack)
{
    int i = blockIdx.x * blockDim.x + threadIdx.x;
    if (i < NR_K - 1) {
        rad_pack[2 * i]     = make_float4(rad_a[2 * i], rad_a[2 * i + 1],
                                          rad_b[2 * i], rad_b[2 * i + 1]);
        rad_pack[2 * i + 1] = make_float4(rad_c[2 * i], rad_c[2 * i + 1],
                                          rad_d[2 * i], rad_d[2 * i + 1]);
    }
    if (i < NRHO_K - 1) {
        emb_pack[i] = make_float4(emb_a[i], emb_b[i], emb_c[i], emb_d[i]);
    }
}

// Zero E + forces (d_out) and node_rho + Fprime (workspace).
__global__ __launch_bounds__(256) void eam_zero(float* __restrict__ out, int n_out,
                                                float* __restrict__ ws, int n_ws) {
    int i = blockIdx.x * blockDim.x + threadIdx.x;
    if (i < n_out) out[i] = 0.0f;
    if (i < n_ws)  ws[i]  = 0.0f;
}

// ===================== unpacked fallback kernels ============================

__global__ __launch_bounds__(256) void eam_pass1(
    const float* __restrict__ r, const int* __restrict__ dst,
    const float* __restrict__ rs,
    const float* __restrict__ rad_a, const float* __restrict__ rad_b,
    const float* __restrict__ rad_c, const float* __restrict__ rad_d,
    float* __restrict__ node_rho, float* __restrict__ E, int n_edges)
{
    const float inv_h = (float)(NR_K - 1) / CUTOFF_F;
    const int stride = gridDim.x * blockDim.x;
    const float3* __restrict__ r3 = reinterpret_cast<const float3*>(r);

    float phi_acc = 0.0f;
    for (int e = blockIdx.x * blockDim.x + threadIdx.x; e < n_edges; e += stride) {
        float3 v = r3[e];
        float bl = sqrtf(fmaf(v.x, v.x, fmaf(v.y, v.y, v.z * v.z)));
        int idx = min((int)(bl * inv_h), NR_K - 2);
        float s = bl - rs[idx];
        float a0 = rad_a[2*idx], a1 = rad_a[2*idx+1];
        float b0 = rad_b[2*idx], b1 = rad_b[2*idx+1];
        float c0 = rad_c[2*idx], c1 = rad_c[2*idx+1];
        float d0 = rad_d[2*idx], d1 = rad_d[2*idx+1];
        float rho  = fmaf(s, fmaf(s, fmaf(s, d0, c0), b0), a0);
        float rphi = fmaf(s, fmaf(s, fmaf(s, d1, c1), b1), a1);
        atomic_add_f32(&node_rho[dst[e]], rho);
        phi_acc += 0.5f * rphi / bl;
    }
    float w = wave_sum(phi_acc);
    if ((threadIdx.x & 31) == 0) atomic_add_f32(E, w);
}

__global__ __launch_bounds__(256) void eam_pass2(
    const float* __restrict__ node_rho, const float* __restrict__ rhos,
    const float* __restrict__ emb_a, const float* __restrict__ emb_b,
    const float* __restrict__ emb_c, const float* __restrict__ emb_d,
    float* __restrict__ Fprime, float* __restrict__ E, int n_nodes)
{
    const float inv_h = (float)(NRHO_K - 1) / RHOMAX_F;
    int i = blockIdx.x * blockDim.x + threadIdx.x;
    float Facc = 0.0f;
    if (i < n_nodes) {
        float rho = node_rho[i];
        int idx = max(0, min((int)(rho * inv_h), NRHO_K - 2));
        float s = rho - rhos[idx];
        float a = emb_a[idx], b = emb_b[idx], c = emb_c[idx], d = emb_d[idx];
        Facc      = fmaf(s, fmaf(s, fmaf(s, d, c), b), a);
        Fprime[i] = fmaf(s, fmaf(s, 3.0f * d, 2.0f * c), b);
    }
    float w = wave_sum(Facc);
    if ((threadIdx.x & 31) == 0) atomic_add_f32(E, w);
}

__global__ __launch_bounds__(256) void eam_pass3(
    const float* __restrict__ r, const int* __restrict__ src, const int* __restrict__ dst,
    const float* __restrict__ rs,
    const float* __restrict__ rad_a, const float* __restrict__ rad_b,
    const float* __restrict__ rad_c, const float* __restrict__ rad_d,
    const float* __restrict__ Fprime, float* __restrict__ forces, int n_edges)
{
    const float inv_h = (float)(NR_K - 1) / CUTOFF_F;
    const int stride = gridDim.x * blockDim.x;
    const float3* __restrict__ r3 = reinterpret_cast<const float3*>(r);

    for (int e = blockIdx.x * blockDim.x + threadIdx.x; e < n_edges; e += stride) {
        float3 v = r3[e];
        float bl = sqrtf(fmaf(v.x, v.x, fmaf(v.y, v.y, v.z * v.z)));
        int idx = min((int)(bl * inv_h), NR_K - 2);
        float s = bl - rs[idx];
        float b0 = rad_b[2*idx], c0 = rad_c[2*idx], d0 = rad_d[2*idx];
        float a1 = rad_a[2*idx+1], b1 = rad_b[2*idx+1];
        float c1 = rad_c[2*idx+1], d1 = rad_d[2*idx+1];
        float drho  = fmaf(s, fmaf(s, 3.0f * d0, 2.0f * c0), b0);
        float rphi  = fmaf(s, fmaf(s, fmaf(s, d1, c1), b1), a1);
        float rphip = fmaf(s, fmaf(s, 3.0f * d1, 2.0f * c1), b1);
        float inv_bl = 1.0f / bl;
        float dphi = (rphip - rphi * inv_bl) * inv_bl;
        int de = dst[e], se = src[e];
        float g = fmaf(Fprime[de], drho, 0.5f * dphi);
        float scale = g * inv_bl;
        float fx = scale * v.x, fy = scale * v.y, fz = scale * v.z;
        atomic_add_f32(&forces[3 * se + 0],  fx);
        atomic_add_f32(&forces[3 * se + 1],  fy);
        atomic_add_f32(&forces[3 * se + 2],  fz);
        atomic_add_f32(&forces[3 * de + 0], -fx);
        atomic_add_f32(&forces[3 * de + 1], -fy);
        atomic_add_f32(&forces[3 * de + 2], -fz);
    }
}

extern "C" void kernel_launch(void* const* d_in, const int* in_sizes, int n_in,
                              void* d_out, int out_size, void* d_ws, size_t ws_size,
                              hipStream_t stream) {
    // setup_inputs() order:
    // 0:r 1:src 2:dst 3:n_nodes 4:rs 5:rhos 6..9:rad_a..d 10..13:emb_a..d
    const float* r     = (const float*)d_in[0];
    const int*   src   = (const int*)  d_in[1];
    const int*   dst   = (const int*)  d_in[2];
    const float* rs    = (const float*)d_in[4];
    const float* rhos  = (const float*)d_in[5];
    const float* rad_a = (const float*)d_in[6];
    const float* rad_b = (const float*)d_in[7];
    const float* rad_c = (const float*)d_in[8];
    const float* rad_d = (const float*)d_in[9];
    const float* emb_a = (const float*)d_in[10];
    const float* emb_b = (const float*)d_in[11];
    const float* emb_c = (const float*)d_in[12];
    const float* emb_d = (const float*)d_in[13];

    const int n_edges = in_sizes[0] / 3;
    const int n_nodes = (out_size - 1) / 3;   // out = [E, forces(n_nodes*3)]

    float* E      = (float*)d_out;
    float* forces = E + 1;

    const size_t rad_bytes  = (size_t)(NR_K - 1)  * 8 * sizeof(float);   // 32B/rec
    const size_t emb_bytes  = (size_t)(NRHO_K - 1) * 4 * sizeof(float);  // 16B/rec
    const size_t node_bytes = (size_t)n_nodes * sizeof(float);
    const bool packed = ws_size >= rad_bytes + emb_bytes + 2 * node_bytes;

    float4* rad_pack = nullptr;
    float4* emb_pack = nullptr;
    float*  node_rho;
    float*  Fprime;
    if (packed) {
        char* p  = (char*)d_ws;
        rad_pack = (float4*)p;              p += rad_bytes;
        emb_pack = (float4*)p;              p += emb_bytes;
        node_rho = (float*)p;               p += node_bytes;
        Fprime   = (float*)p;
    } else {
        node_rho = (float*)d_ws;
        Fprime   = node_rho + n_nodes;
    }
    const int n_ws_elems = 2 * n_nodes;

    const int T = 256;
    int n_zero = out_size > n_ws_elems ? out_size : n_ws_elems;
    eam_zero<<<(n_zero + T - 1) / T, T, 0, stream>>>(E, out_size, node_rho, n_ws_elems);

    // grid-stride: ~4 edges per thread so the software prefetch has a target
    int blocks_e = (n_edges + T * 4 - 1) / (T * 4);
    if (blocks_e < 1) blocks_e = 1;
    int blocks_n = (n_nodes + T - 1) / T;

    if (packed) {
        int k = (NR_K > NRHO_K ? NR_K : NRHO_K) - 1;
        eam_repack<<<(k + T - 1) / T, T, 0, stream>>>(rad_a, rad_b, rad_c, rad_d,
                                                      emb_a, emb_b, emb_c, emb_d,
                                                      rad_pack, emb_pack);
        eam_pass1_packed<<<blocks_e, T, 0, stream>>>(r, dst, rs, rad_pack,
                                                     node_rho, E, n_edges);
        eam_pass2_packed<<<blocks_n, T, 0, stream>>>(node_rho, rhos, emb_pack,
                                                     Fprime, E, n_nodes);
        eam_pass3_packed<<<blocks_e, T, 0, stream>>>(r, src, dst, rs, rad_pack,
                                                     Fprime, forces, n_edges);
    } else {
        eam_pass1<<<blocks_e, T, 0, stream>>>(r, dst, rs, rad_a, rad_b, rad_c, rad_d,
                                              node_rho, E, n_edges);
        eam_pass2<<<blocks_n, T, 0, stream>>>(node_rho, rhos,
                                              emb_a, emb_b, emb_c, emb_d,
                                              Fprime, E, n_nodes);
        eam_pass3<<<blocks_e, T, 0, stream>>>(r, src, dst, rs, rad_a, rad_b, rad_c, rad_d,
                                              Fprime, forces, n_edges);
    }
}